// ImageProcessing_39986145526222
// MI455X (gfx1250) — compile-verified
//
#include <hip/hip_runtime.h>

// Fused flip -> rotate(nearest) -> bilinear resized-crop -> normalize.
// Flips and crop offsets are folded algebraically into the inverse-rotation
// affine, so each bilinear corner costs ~8 VALU ops. 4 output pixels per
// thread (shared y-axis math, b128 nontemporal stores). Memory-bound:
// single pass over input (<=201 MB) + NT-store output (38.5 MB) ~= 10us floor.

typedef float v4f __attribute__((ext_vector_type(4)));

namespace {

constexpr int BATCH = 64;
constexpr int CHANS = 3;
constexpr int IMG_H = 512;
constexpr int IMG_W = 512;
constexpr int OUT   = 224;
constexpr int NPIX  = OUT * OUT;            // 50176
constexpr int PXT   = 4;                    // pixels per thread (along x)
constexpr int TPB   = 256;                  // 8 wave32s
constexpr int COLS4 = OUT / PXT;            // 56 threads per output row
constexpr int T_PER_IMG   = NPIX / PXT;     // 12544
constexpr int BLK_PER_IMG = T_PER_IMG / TPB; // 49 (exact)

// One bilinear corner: rounded source coords (flip already folded into the
// affine), in-frame test, flat offset via single exact float FMA.
__device__ __forceinline__ void corner(float gx, float gy, float bw,
                                       int& off, float& w) {
    float rx = rintf(gx);                   // v_rndne: round-half-even == jnp.round
    float ry = rintf(gy);
    bool v = (rx >= 0.0f) & (rx <= 511.0f) & (ry >= 0.0f) & (ry <= 511.0f);
    float of = fmaf(ry, 512.0f, rx);        // <= 2^18: exact in f32
    int   o  = (int)of;
    off = v ? o : 0;                        // invalid -> harmless read of base[0]
    w   = v ? bw : 0.0f;                    // ...with zero weight (fill=0)
}

} // namespace

__global__ __launch_bounds__(TPB) void fused_augment_kernel(
    const float* __restrict__ img,        // [B,C,H,W]
    const float* __restrict__ angle,      // [B]
    const float* __restrict__ mean,       // [C]
    const float* __restrict__ stdv,       // [C]
    const int*   __restrict__ hflip,      // [B]
    const int*   __restrict__ vflip,      // [B]
    const int*   __restrict__ crop_top,   // [B]
    const int*   __restrict__ crop_left,  // [B]
    const int*   __restrict__ crop_h,     // [B]
    const int*   __restrict__ crop_w,     // [B]
    float*       __restrict__ out)        // [B,C,OUT,OUT]
{
    const int b   = blockIdx.y;                         // wave-uniform image id
    const int t   = blockIdx.x * TPB + threadIdx.x;     // 0 .. T_PER_IMG-1
    const int r   = t / COLS4;                          // output row
    const int oxb = (t - r * COLS4) * PXT;              // first of 4 output cols

    // ---------------- uniform per-image parameters (scalar path) ----------
    const float theta = angle[b] * 0.017453292519943295f;   // <= 0.1746 rad
    const float t2 = theta * theta;
    // Taylor sin/cos: exact to f32 over [0, 0.175] -- no libm, ~6 uniform FMAs
    const float sn = theta * fmaf(t2, fmaf(t2, 8.3333333e-3f, -0.16666667f), 1.0f);
    const float cs = fmaf(t2, fmaf(t2, fmaf(t2, -1.3888889e-3f, 4.1666668e-2f),
                                   -0.5f), 1.0f);
    const int hf = hflip[b];
    const int vf = vflip[b];
    const int ctop  = crop_top[b];
    const int cleft = crop_left[b];
    const int chh   = crop_h[b];
    const int cww   = crop_w[b];

    // Fold flips into the inverse-rotation coefficients (reflection about the
    // 255.5 center is a sign flip: 511 - (a*x + b*y + 255.5) = -a*x - b*y + 255.5)
    const float axc = hf ? -cs : cs;
    const float ayc = hf ? -sn : sn;
    const float yxc = vf ?  sn : -sn;
    const float yyc = vf ? -cs : cs;
    // Fold crop origin + image center into uniform constants:
    //   gx = axc*xloc + ayc*yloc + Kx   (xloc,yloc are crop-local coords)
    const float CX = (float)cleft - 255.5f;
    const float CY = (float)ctop  - 255.5f;
    const float Kx = fmaf(axc, CX, fmaf(ayc, CY, 255.5f));
    const float Ky = fmaf(yxc, CX, fmaf(yyc, CY, 255.5f));

    const float inv_out = 1.0f / (float)OUT;
    const float shf  = (float)chh * inv_out;
    const float swf  = (float)cww * inv_out;
    const float chm1 = (float)(chh - 1);
    const float cwm1 = (float)(cww - 1);

    // ---------------- y-axis coords: shared by the 4 pixels ----------------
    float syf = fmaxf(fmaf((float)r + 0.5f, shf, -0.5f), 0.0f);
    float y0f = floorf(syf);
    float wy  = syf - y0f;
    float y1f = fminf(y0f + 1.0f, chm1);
    float dyf = y1f - y0f;                  // 0 or 1

    int   off[PXT][4];
    float w  [PXT][4];

#pragma unroll
    for (int p = 0; p < PXT; ++p) {
        float sxf = fmaxf(fmaf((float)(oxb + p) + 0.5f, swf, -0.5f), 0.0f);
        float x0f = floorf(sxf);
        float wx  = sxf - x0f;
        float x1f = fminf(x0f + 1.0f, cwm1);
        float dxf = x1f - x0f;              // 0 or 1

        // corner 00 via the folded affine; 01/10/11 via 2-FMA deltas
        float gx00 = fmaf(axc, x0f, fmaf(ayc, y0f, Kx));
        float gy00 = fmaf(yxc, x0f, fmaf(yyc, y0f, Ky));
        float gx01 = fmaf(axc, dxf, gx00);
        float gy01 = fmaf(yxc, dxf, gy00);
        float gx10 = fmaf(ayc, dyf, gx00);
        float gy10 = fmaf(yyc, dyf, gy00);
        float gx11 = fmaf(ayc, dyf, gx01);
        float gy11 = fmaf(yyc, dyf, gy01);

        const float bw11 = wy * wx;
        const float bw10 = wy - bw11;            // wy*(1-wx)
        const float bw01 = wx - bw11;            // (1-wy)*wx
        const float bw00 = 1.0f - wy - bw01;     // (1-wy)*(1-wx)

        corner(gx00, gy00, bw00, off[p][0], w[p][0]);
        corner(gx01, gy01, bw01, off[p][1], w[p][1]);
        corner(gx10, gy10, bw10, off[p][2], w[p][2]);
        corner(gx11, gy11, bw11, off[p][3], w[p][3]);
    }

    // ---------------- gather + blend + normalize, per channel --------------
    const float* __restrict__ imb = img + (size_t)b * CHANS * (IMG_H * IMG_W);
    float* outp = out + (size_t)b * CHANS * NPIX + (size_t)r * OUT + oxb;

#pragma unroll
    for (int c = 0; c < CHANS; ++c) {
        const float* __restrict__ base = imb + (size_t)c * (IMG_H * IMG_W);
        // (v/255 - mean)/std == v*(1/(255*std)) + (-mean/std)  (uniform)
        const float sc = 1.0f / (255.0f * stdv[c]);
        const float bi = -mean[c] / stdv[c];

        v4f res;
#pragma unroll
        for (int p = 0; p < PXT; ++p) {
            float acc =      w[p][0] * base[off[p][0]];
            acc = fmaf(w[p][1], base[off[p][1]], acc);
            acc = fmaf(w[p][2], base[off[p][2]], acc);
            acc = fmaf(w[p][3], base[off[p][3]], acc);
            res[p] = fmaf(acc, sc, bi);
        }
        // write-once output: nontemporal b128 store (TH_STORE_NT) keeps the
        // 192 MB L2 free for the input working set
        __builtin_nontemporal_store(res, (v4f*)(outp + (size_t)c * NPIX));
    }
}

extern "C" void kernel_launch(void* const* d_in, const int* in_sizes, int n_in,
                              void* d_out, int out_size, void* d_ws, size_t ws_size,
                              hipStream_t stream) {
    (void)in_sizes; (void)n_in; (void)out_size; (void)d_ws; (void)ws_size;

    const float* img       = (const float*)d_in[0];
    const float* angle     = (const float*)d_in[1];
    const float* mean      = (const float*)d_in[2];
    const float* stdv      = (const float*)d_in[3];
    const int*   hflip     = (const int*)d_in[4];
    const int*   vflip     = (const int*)d_in[5];
    const int*   crop_top  = (const int*)d_in[6];
    const int*   crop_left = (const int*)d_in[7];
    const int*   crop_h    = (const int*)d_in[8];
    const int*   crop_w    = (const int*)d_in[9];
    float*       out       = (float*)d_out;

    dim3 grid(BLK_PER_IMG, BATCH);   // 49 x 64 blocks
    dim3 block(TPB);                 // 256 threads = 8 wave32s
    fused_augment_kernel<<<grid, block, 0, stream>>>(
        img, angle, mean, stdv, hflip, vflip,
        crop_top, crop_left, crop_h, crop_w, out);
}